// HMM_30855045055177
// MI455X (gfx1250) — compile-verified
//
#include <hip/hip_runtime.h>

// HMM forward-backward smoothing, MI455X (gfx1250, wave32).
//   B=256, T=4096, K=32.  Output [B,T,K] f32.
//
// Linear-domain recursions (per-(b,t) log offsets cancel in the posterior
// normalization) with lossless power-of-2 per-column rescaling.  Each of the
// 4096 dependent steps is a 32x32 fp32 matvec over a 16-wide batch tile:
// 16x V_WMMA_F32_16X16X4_F32 in 4 independent accumulator chains (depth 4).
// Emissions evaluated as a log2-domain Horner polynomial feeding raw
// v_exp_f32 / v_log_f32 (arguments provably normal-range, no fixups needed).
// 16 forward chains write ln_f to d_out, 16 backward chains write ln_b to
// d_ws (128 MiB), then a memory-bound combine kernel normalizes over K.

#define K_STATES 32
#define T_LEN    4096
#define BATCH_N  256
#define LOG2E    1.4426950408889634f
#define LN2      0.6931471805599453f

typedef __attribute__((ext_vector_type(2))) float v2f;
typedef __attribute__((ext_vector_type(4))) float v4f;
typedef __attribute__((ext_vector_type(8))) float v8f;

// per-batch-column max over all 32 states: depth-4 tree + lane-half swap
__device__ __forceinline__ float colmax(const float (&v)[16]) {
  float a[8];
#pragma unroll
  for (int i = 0; i < 8; ++i) a[i] = fmaxf(v[i], v[i + 8]);
#pragma unroll
  for (int s = 4; s >= 1; s >>= 1)
#pragma unroll
    for (int i = 0; i < s; ++i) a[i] = fmaxf(a[i], a[i + s]);
  return fmaxf(a[0], __shfl_xor(a[0], 16, 32));
}

// exact power-of-two reciprocal scale (any positive scale is valid; pow2
// keeps the rescale bit-lossless)
__device__ __forceinline__ float rcp_pow2(float m) {
  int e = (__float_as_int(m) >> 23) & 0xff;
  return __int_as_float((254 - e) << 23);
}

// Relayout 32x16 from WMMA C-layout (float c[16]: regs 0-7 = states (r&7)+8h,
// regs 8-15 = 16+(r&7)+8h for lane-half h) into WMMA B-layout
// (v2f fB[8]: fB[ki].{x,y} = state 4ki+2h+{0,1}).  Value (k,n) lives in
// lane-half (k>>3)&1 at reg (k&7)+8*(k>>4): one uniform half-swap + one
// cndmask per element.
__device__ __forceinline__ void c_to_b(const float (&c)[16], v2f (&fB)[8], int hi) {
#pragma unroll
  for (int ki = 0; ki < 8; ++ki) {
#pragma unroll
    for (int j = 0; j < 2; ++j) {
      const int kt0 = 4 * ki + j;                   // state wanted by h=0 lanes
      const int r0  = (kt0 & 7) + ((ki >> 2) << 3); // its reg index
      const int r1  = r0 + 2;                       // reg index for h=1 lanes
      const int sh  = (kt0 >> 3) & 1;               // source lane-half
      float shl = __shfl_xor(sh ? c[r0] : c[r1], 16, 32);
      float val = sh ? (hi ? c[r1] : shl) : (hi ? shl : c[r0]);
      if (j == 0) fB[ki].x = val; else fB[ki].y = val;
    }
  }
}

// store one (b,t) row of 32 floats from C-layout: lane-half h covers
// k = 8h..8h+7 and 16+8h..16+8h+7, as 4 streaming b128 stores per lane
__device__ __forceinline__ void store_row(float* __restrict__ row,
                                          const float (&v)[16], int hi) {
  v4f w0 = {v[0],  v[1],  v[2],  v[3]};
  v4f w1 = {v[4],  v[5],  v[6],  v[7]};
  v4f w2 = {v[8],  v[9],  v[10], v[11]};
  v4f w3 = {v[12], v[13], v[14], v[15]};
  __builtin_nontemporal_store(w0, (v4f*)(row +      8 * hi));
  __builtin_nontemporal_store(w1, (v4f*)(row +      8 * hi + 4));
  __builtin_nontemporal_store(w2, (v4f*)(row + 16 + 8 * hi));
  __builtin_nontemporal_store(w3, (v4f*)(row + 16 + 8 * hi + 4));
}

// g = A(32x32) @ f(32x16): 2 M-tiles x 8 K-tiles of V_WMMA_F32_16X16X4_F32,
// split into 4 independent accumulator chains (depth 4) + pk recombine
__device__ __forceinline__ void gemm32(const v2f (&A)[2][8], const v2f (&Bm)[8],
                                       float (&g)[16]) {
  v8f a0 = {}, a1 = {}, b0 = {}, b1 = {};
#pragma unroll
  for (int ki = 0; ki < 4; ++ki) {
    a0 = __builtin_amdgcn_wmma_f32_16x16x4_f32(false, A[0][ki],     false, Bm[ki],
                                               (short)0, a0, false, false);
    a1 = __builtin_amdgcn_wmma_f32_16x16x4_f32(false, A[0][ki + 4], false, Bm[ki + 4],
                                               (short)0, a1, false, false);
    b0 = __builtin_amdgcn_wmma_f32_16x16x4_f32(false, A[1][ki],     false, Bm[ki],
                                               (short)0, b0, false, false);
    b1 = __builtin_amdgcn_wmma_f32_16x16x4_f32(false, A[1][ki + 4], false, Bm[ki + 4],
                                               (short)0, b1, false, false);
  }
#pragma unroll
  for (int r = 0; r < 8; ++r) { g[r] = a0[r] + a1[r]; g[8 + r] = b0[r] + b1[r]; }
}

__global__ __launch_bounds__(32) void hmm_scan(
    const float* __restrict__ obvs,      // [256][4096]
    const float* __restrict__ log_init,  // [32]
    const float* __restrict__ trans,     // [32][32], strictly positive
    const float* __restrict__ means,     // [32]
    const float* __restrict__ log_sig,   // [32]
    float* __restrict__ lnf,             // d_out: [256][4096][32]
    float* __restrict__ lnb)             // d_ws : [256][4096][32]
{
  const int L   = threadIdx.x;
  const int hi  = L >> 4;           // lane half
  const int n   = L & 15;           // batch column within tile / matrix row
  const int dir = blockIdx.x >> 4;  // 0 = forward, 1 = backward
  const int b   = ((blockIdx.x & 15) << 4) + n;
  const size_t obase = (size_t)b * T_LEN;
  const size_t rbase = obase * K_STATES;

  // transition matrix in WMMA A-layout, resident for the whole chain
  v2f A[2][8];
#pragma unroll
  for (int mi = 0; mi < 2; ++mi)
#pragma unroll
    for (int ki = 0; ki < 8; ++ki) {
      int m  = mi * 16 + n;
      int ka = ki * 4 + 2 * hi;
      A[mi][ki].x = trans[m * 32 + ka];
      A[mi][ki].y = trans[m * 32 + ka + 1];
    }

  if (dir == 0) {
    // ---------------- forward: f[t] = e[t] .* (A f[t-1]) ----------------
    // log2-domain emission Horner coeffs at C-layout states
    // kC(r) = (r&7) + 8h + 16*(r>>3):
    //   log2 e(x,k) = pa*x^2 + pb*x + pc
    float pa[16], pb[16], pc[16];
#pragma unroll
    for (int r = 0; r < 16; ++r) {
      int k = (r & 7) + 8 * hi + ((r >> 3) << 4);
      float ls = log_sig[k];
      float mu = means[k];
      float a  = -0.5f * __expf(-2.0f * ls) * LOG2E;
      pa[r] = a;
      pb[r] = -2.0f * mu * a;
      pc[r] = mu * mu * a + (-ls - 0.91893853320467274f) * LOG2E;
    }

    v2f fB[8];
    { // t = 0: ln f0 = ln_e0 + log_init (exact), then normalized linear f0
      float x = obvs[obase];
      float lo[16];
#pragma unroll
      for (int r = 0; r < 16; ++r) {
        int k = (r & 7) + 8 * hi + ((r >> 3) << 4);
        float le2 = fmaf(x, fmaf(x, pa[r], pb[r]), pc[r]);
        lo[r] = fmaf(le2, LN2, log_init[k]);
      }
      store_row(lnf + rbase, lo, hi);
      float m = colmax(lo);
      float c2[16];
#pragma unroll
      for (int r = 0; r < 16; ++r)
        c2[r] = __builtin_amdgcn_exp2f((lo[r] - m) * LOG2E);
      c_to_b(c2, fB, hi);
    }

    float xnext = obvs[obase + 1];
    for (int t = 1; t < T_LEN; ++t) {
      float x = xnext;
      int tn = (t + 1 < T_LEN) ? (t + 1) : (T_LEN - 1);
      xnext = obvs[obase + tn];                    // prefetch next observation

      float g[16];
      gemm32(A, fB, g);                            // g = A @ f_prev

      float le2[16], hv[16], lo[16];
#pragma unroll
      for (int r = 0; r < 16; ++r) {
        le2[r] = fmaf(x, fmaf(x, pa[r], pb[r]), pc[r]);     // log2 e[t,k]
        hv[r]  = g[r] * __builtin_amdgcn_exp2f(le2[r]);     // linear f
        // stored natural log (per-(b,t) offset cancels later)
        lo[r]  = (__builtin_amdgcn_logf(g[r]) + le2[r]) * LN2;
      }
      store_row(lnf + rbase + (size_t)t * K_STATES, lo, hi);

      float s = rcp_pow2(colmax(hv));              // lossless pow2 rescale
#pragma unroll
      for (int r = 0; r < 16; ++r) hv[r] *= s;
      c_to_b(hv, fB, hi);
    }
  } else {
    // ------------- backward: b[t] = A (b[t+1] .* e[t+1]) -------------
    // coeffs at B-layout states kB(q) = 4*(q>>1) + 2h + (q&1)
    float pa[16], pb[16], pc[16];
    v2f fB[8];
#pragma unroll
    for (int q = 0; q < 16; ++q) {
      int k = ((q >> 1) << 2) + 2 * hi + (q & 1);
      float ls = log_sig[k];
      float mu = means[k];
      float a  = -0.5f * __expf(-2.0f * ls) * LOG2E;
      pa[q] = a;
      pb[q] = -2.0f * mu * a;
      pc[q] = mu * mu * a + (-ls - 0.91893853320467274f) * LOG2E;
      float v = __expf(log_init[k]);               // b[T-1] = exp(log_init)
      if (q & 1) fB[q >> 1].y = v; else fB[q >> 1].x = v;
    }
    { // store ln_b[T-1] = log_init (C-layout addresses)
      float lo[16];
#pragma unroll
      for (int r = 0; r < 16; ++r) {
        int k = (r & 7) + 8 * hi + ((r >> 3) << 4);
        lo[r] = log_init[k];
      }
      store_row(lnb + rbase + (size_t)(T_LEN - 1) * K_STATES, lo, hi);
    }

    float xnext = obvs[obase + (T_LEN - 1)];
    for (int t = T_LEN - 2; t >= 0; --t) {
      float x = xnext;                             // obvs[b, t+1]
      int tn = (t > 0) ? t : 1;
      xnext = obvs[obase + tn];

      v2f uB[8];                                   // u = b[t+1] .* e[t+1]
#pragma unroll
      for (int q = 0; q < 16; ++q) {
        float le2 = fmaf(x, fmaf(x, pa[q], pb[q]), pc[q]);
        float e   = __builtin_amdgcn_exp2f(le2);
        float f   = (q & 1) ? fB[q >> 1].y : fB[q >> 1].x;
        float u   = f * e;
        if (q & 1) uB[q >> 1].y = u; else uB[q >> 1].x = u;
      }

      float g[16];
      gemm32(A, uB, g);                            // b[t] = A @ u

      float lo[16];
#pragma unroll
      for (int r = 0; r < 16; ++r)
        lo[r] = __builtin_amdgcn_logf(g[r]) * LN2;
      store_row(lnb + rbase + (size_t)t * K_STATES, lo, hi);

      float s = rcp_pow2(colmax(g));
#pragma unroll
      for (int r = 0; r < 16; ++r) g[r] *= s;
      c_to_b(g, fB, hi);
    }
  }
}

// out = (lnf + lnb) - logsumexp_K(lnf + lnb); fully coalesced, 4 KB per wave
__global__ __launch_bounds__(256) void hmm_combine(const float* __restrict__ lnb,
                                                   float* __restrict__ io) {
  size_t idx = (size_t)blockIdx.x * 256 + threadIdx.x;   // over B*T
  float* p = io + idx * K_STATES;
  const float* q = lnb + idx * K_STATES;
  float s[32];
#pragma unroll
  for (int i = 0; i < 8; ++i) {
    float4 a = *(const float4*)(p + 4 * i);
    float4 c = *(const float4*)(q + 4 * i);
    s[4 * i + 0] = a.x + c.x; s[4 * i + 1] = a.y + c.y;
    s[4 * i + 2] = a.z + c.z; s[4 * i + 3] = a.w + c.w;
  }
  float m[16];
#pragma unroll
  for (int i = 0; i < 16; ++i) m[i] = fmaxf(s[i], s[i + 16]);
#pragma unroll
  for (int st = 8; st >= 1; st >>= 1)
#pragma unroll
    for (int i = 0; i < st; ++i) m[i] = fmaxf(m[i], m[i + st]);
  float mx = m[0];
  float sum = 0.0f;
#pragma unroll
  for (int i = 0; i < 32; ++i)
    sum += __builtin_amdgcn_exp2f((s[i] - mx) * LOG2E);
  float lse = fmaf(__builtin_amdgcn_logf(sum), LN2, mx);
#pragma unroll
  for (int i = 0; i < 8; ++i) {
    *(float4*)(p + 4 * i) = make_float4(s[4 * i + 0] - lse, s[4 * i + 1] - lse,
                                        s[4 * i + 2] - lse, s[4 * i + 3] - lse);
  }
}

extern "C" void kernel_launch(void* const* d_in, const int* in_sizes, int n_in,
                              void* d_out, int out_size, void* d_ws, size_t ws_size,
                              hipStream_t stream) {
  (void)in_sizes; (void)n_in; (void)out_size; (void)ws_size;
  const float* obvs     = (const float*)d_in[0];
  const float* log_init = (const float*)d_in[1];
  const float* trans    = (const float*)d_in[2];
  const float* means    = (const float*)d_in[3];
  const float* log_sig  = (const float*)d_in[4];
  float* out = (float*)d_out;
  float* lnb = (float*)d_ws;  // needs 256*4096*32*4 = 128 MiB scratch

  // 16 forward + 16 backward single-wave chains
  hmm_scan<<<dim3(32), dim3(32), 0, stream>>>(obvs, log_init, trans, means,
                                              log_sig, out, lnb);
  // normalize: (B*T)/256 = 4096 blocks
  hmm_combine<<<dim3((BATCH_N * T_LEN) / 256), dim3(256), 0, stream>>>(lnb, out);
}